// DynamicMultiLinear_8589934789
// MI455X (gfx1250) — compile-verified
//
#include <hip/hip_runtime.h>

// DynamicMultiLinear on MI455X (gfx1250):
//   y[row] = x[row] @ W[g]^T + b[g],  rows grouped into contiguous segments per g.
// fp32 path via V_WMMA_F32_16X16X4_F32 (matches reference precision; ~57 FLOP/byte
// puts the problem near the 23.3 TB/s HBM roofline, so fp32 matrix rate suffices).
// This revision: 32x64 per-wave tile (B fragments reused across 2 WMMAs, -40% LDS
// fragment traffic) + register-prefetch pipeline hiding global->LDS latency.

#define G_NETS   256
#define IN_F     256
#define OUT_F    256
#define NROWS    262144
#define PAD      2048

#define TILE_M   64          // rows per block
#define KC       32          // K-chunk staged in LDS
#define XPITCH   36          // padded LDS pitch (floats): conflict-free b64 frags,
#define WPITCH   36          // 16B-aligned b128 chunk stores (36*4 = 144 % 16 == 0)

typedef float v2f __attribute__((ext_vector_type(2)));
typedef float v4f __attribute__((ext_vector_type(4)));
typedef float v8f __attribute__((ext_vector_type(8)));

// ---------------------------------------------------------------------------
// Kernel 1: exclusive prefix sum of counts -> segment offsets (G=256, 1 block)
// ---------------------------------------------------------------------------
__global__ void dml_scan_offsets(const int* __restrict__ counts,
                                 int* __restrict__ offs) {
    __shared__ int tmp[G_NETS];
    int t = threadIdx.x;
    int c = counts[t];
    tmp[t] = c;
    __syncthreads();
    for (int d = 1; d < G_NETS; d <<= 1) {   // Hillis-Steele inclusive scan
        int v = (t >= d) ? tmp[t - d] : 0;
        __syncthreads();
        tmp[t] += v;
        __syncthreads();
    }
    offs[t] = tmp[t] - c;   // exclusive
}

// ---------------------------------------------------------------------------
// Kernel 2: grouped GEMM. grid=(G, PAD/TILE_M), block=256 (8 wave32s, 2x4).
// Block: 64 rows x 256 cols of one segment. Wave: 32 rows x 64 cols
// = 8 accumulators of 16x16; each B fragment feeds 2 WMMAs.
// ---------------------------------------------------------------------------
__global__ __launch_bounds__(256)
void dml_grouped_gemm(const float* __restrict__ weight,   // [G, OUT_F, IN_F]
                      const float* __restrict__ bias,     // [G, OUT_F]
                      const float* __restrict__ x,        // [N, IN_F]
                      const int*   __restrict__ counts,   // [G]
                      const int*   __restrict__ offs,     // [G] exclusive
                      float*       __restrict__ out) {    // [N, OUT_F]
    const int g    = blockIdx.x;
    const int tile = blockIdx.y;
    const int cnt  = counts[g];
    const int row0 = tile * TILE_M;            // row offset within segment
    if (row0 >= cnt) return;                   // ragged grid: dead tile
    const int seg_base = offs[g];

    __shared__ float xsh[TILE_M * XPITCH];     //  9.2 KB
    __shared__ float wsh[OUT_F * WPITCH];      // 36.9 KB  (46 KB total)

    const int t      = threadIdx.x;
    const int lane   = t & 31;
    const int wave   = t >> 5;
    const int wave_m = wave >> 2;              // 0..1 -> rows  wave_m*32
    const int wave_n = wave & 3;               // 0..3 -> cols  wave_n*64
    const int l15    = lane & 15;
    const int hi     = lane >> 4;              // selects K pair / M+8 half

    const float* wg = weight + (size_t)g * OUT_F * IN_F;

    // Init accumulators with bias broadcast.
    // C/D layout: lane 0-15 -> N=lane, VGPR v -> M=v (lanes<16) / v+8 (lanes>=16).
    v8f acc0[4], acc1[4];
#pragma unroll
    for (int j = 0; j < 4; ++j) {
        float bv = bias[g * OUT_F + wave_n * 64 + j * 16 + l15];
#pragma unroll
        for (int v = 0; v < 8; ++v) { acc0[j][v] = bv; acc1[j][v] = bv; }
    }

    // ---- register-prefetch pipeline over K chunks ----
    v4f xp[2];   // X chunk: 64x32 = 512 float4, 2 per thread
    v4f wp[8];   // W chunk: 256x32 = 2048 float4, 8 per thread

    auto fetch = [&](int kc) {
#pragma unroll
        for (int i = 0; i < 2; ++i) {
            int q = i * 256 + t, r = q >> 3, c4 = q & 7;
            int grow = seg_base + row0 + r;     // may spill past segment end
            v4f val = {0.f, 0.f, 0.f, 0.f};
            if (grow < NROWS)
                val = *(const v4f*)(x + (size_t)grow * IN_F + kc + c4 * 4);
            xp[i] = val;
        }
#pragma unroll
        for (int i = 0; i < 8; ++i) {
            int q = i * 256 + t, n = q >> 3, c4 = q & 7;
            wp[i] = *(const v4f*)(wg + (size_t)n * IN_F + kc + c4 * 4);
        }
    };
    auto stage = [&]() {
#pragma unroll
        for (int i = 0; i < 2; ++i) {
            int q = i * 256 + t, r = q >> 3, c4 = q & 7;
            *(v4f*)(xsh + r * XPITCH + c4 * 4) = xp[i];
        }
#pragma unroll
        for (int i = 0; i < 8; ++i) {
            int q = i * 256 + t, n = q >> 3, c4 = q & 7;
            *(v4f*)(wsh + n * WPITCH + c4 * 4) = wp[i];
        }
    };

    fetch(0);
    for (int kc = 0; kc < IN_F; kc += KC) {
        stage();
        __syncthreads();
        if (kc + KC < IN_F) fetch(kc + KC);    // overlap next global loads w/ WMMA

        // A 16x4 fp32 layout: lane m=lane&15; VGPR0/1 hold K pair (2*hi, 2*hi+1).
        // B  4x16 fp32 layout mirrors: lane n=lane&15; K pair (2*hi, 2*hi+1),
        // i.e. W[n][k],W[n][k+1] -- contiguous in the un-transposed LDS chunk.
#pragma unroll
        for (int kb = 0; kb < KC; kb += 4) {
            v2f a0 = *(const v2f*)(xsh + (wave_m * 32 +      l15) * XPITCH + kb + 2 * hi);
            v2f a1 = *(const v2f*)(xsh + (wave_m * 32 + 16 + l15) * XPITCH + kb + 2 * hi);
#pragma unroll
            for (int j = 0; j < 4; ++j) {
                v2f b = *(const v2f*)(wsh + (wave_n * 64 + j * 16 + l15) * WPITCH
                                      + kb + 2 * hi);
                acc0[j] = __builtin_amdgcn_wmma_f32_16x16x4_f32(
                    false, a0, false, b, (short)0, acc0[j], false, false);
                acc1[j] = __builtin_amdgcn_wmma_f32_16x16x4_f32(
                    false, a1, false, b, (short)0, acc1[j], false, false);
            }
        }
        __syncthreads();
    }

    // ---- store (guard ragged tail rows) ----
#pragma unroll
    for (int j = 0; j < 4; ++j) {
        int n = wave_n * 64 + j * 16 + l15;
#pragma unroll
        for (int v = 0; v < 8; ++v) {
            int rb0 = wave_m * 32 + v + 8 * hi;        // acc0 row within block
            int rb1 = rb0 + 16;                         // acc1 row within block
            int sr0 = row0 + rb0, sr1 = row0 + rb1;
            if (sr0 < cnt)
                out[(size_t)(seg_base + sr0) * OUT_F + n] = acc0[j][v];
            if (sr1 < cnt)
                out[(size_t)(seg_base + sr1) * OUT_F + n] = acc1[j][v];
        }
    }
}

// ---------------------------------------------------------------------------
extern "C" void kernel_launch(void* const* d_in, const int* in_sizes, int n_in,
                              void* d_out, int out_size, void* d_ws, size_t ws_size,
                              hipStream_t stream) {
    const float* weight = (const float*)d_in[0];   // [G, OUT_F, IN_F]
    const float* bias   = (const float*)d_in[1];   // [G, 1, OUT_F]
    const float* x      = (const float*)d_in[2];   // [N, IN_F]
    const int*   counts = (const int*)d_in[3];     // [G]
    float*       out    = (float*)d_out;           // [N, OUT_F]
    int*         offs   = (int*)d_ws;              // G ints of scratch

    dml_scan_offsets<<<1, G_NETS, 0, stream>>>(counts, offs);

    dim3 grid(G_NETS, PAD / TILE_M);
    dml_grouped_gemm<<<grid, 256, 0, stream>>>(weight, bias, x, counts, offs, out);
}